// AttentionGRU_39814346834227
// MI455X (gfx1250) — compile-verified
//
#include <hip/hip_runtime.h>
#include <math.h>

typedef __attribute__((ext_vector_type(2))) float v2f;
typedef __attribute__((ext_vector_type(8))) float v8f;

#define HID 64
#define IN  128
#define SEQ 512
#define BATCH 256
#define GATES 192          // 3*HID
#define TILES_PER_BLOCK 8  // M-tiles (16 rows each) per block in kernel 1

static __device__ __forceinline__ v8f wmma_f32(v2f a, v2f b, v8f c) {
    // 8 args: (neg_a, A, neg_b, B, c_mod, C, reuse_a, reuse_b)
    return __builtin_amdgcn_wmma_f32_16x16x4_f32(false, a, false, b, (short)0, c,
                                                 false, false);
}

// CDNA5 async global->LDS copy (16B per lane), tracked by ASYNCcnt.
static __device__ __forceinline__ void async_load_b128(unsigned lds_off,
                                                       const float* gptr) {
    asm volatile("global_load_async_to_lds_b128 %0, %1, off"
                 :: "v"(lds_off), "v"(gptr)
                 : "memory");
}

// ---------------------------------------------------------------------------
// Kernel 1: xg[b*S+s][0..191] = x[b*S+s][:] @ w_ih^T + b_ih
// Grid: (B*S)/(16*TILES_PER_BLOCK) blocks of 128 threads (4 waves).
// Each block walks TILES_PER_BLOCK consecutive 16-row M-tiles; x tiles are
// staged into LDS via double-buffered async b128 copies (8 KB/tile, no
// per-wave redundancy); wave w owns output N-tiles w*3..w*3+2.
// ---------------------------------------------------------------------------
__global__ __launch_bounds__(128) void gru_input_gemm(
    const float* __restrict__ x, const float* __restrict__ w_ih,
    const float* __restrict__ b_ih, float* __restrict__ xg)
{
    __shared__ float sx[2][16][IN];    // two 8 KB x-tile buffers

    const int tid  = threadIdx.x;
    const int lane = tid & 31;
    const int wave = tid >> 5;
    const int mrow = lane & 15;        // A-row / B-col within tile
    const int kh   = (lane >> 4) * 2;  // K sub-offset 0 or 2
    const int n0   = wave * 48;        // first output column for this wave

    const float* w0 = w_ih + (n0 +  0 + mrow) * IN;
    const float* w1 = w_ih + (n0 + 16 + mrow) * IN;
    const float* w2 = w_ih + (n0 + 32 + mrow) * IN;

    const int t0 = blockIdx.x * TILES_PER_BLOCK;

    // Prologue: stage first tile into buffer 0.
    {
        const float* src = x + (long)t0 * (16 * IN);
        const unsigned dst = (unsigned)(size_t)&sx[0][0][0];
#pragma unroll
        for (int p = 0; p < 4; ++p)
            async_load_b128(dst + p * 2048 + tid * 16, src + p * 512 + tid * 4);
    }

    for (int it = 0; it < TILES_PER_BLOCK; ++it) {
        const int cur = it & 1;

        if (it + 1 < TILES_PER_BLOCK) {
            // Prefetch next tile into the other buffer, then wait for current.
            const float* src = x + (long)(t0 + it + 1) * (16 * IN);
            const unsigned dst = (unsigned)(size_t)&sx[cur ^ 1][0][0];
#pragma unroll
            for (int p = 0; p < 4; ++p)
                async_load_b128(dst + p * 2048 + tid * 16,
                                src + p * 512 + tid * 4);
            asm volatile("s_wait_asynccnt 0x4" ::: "memory");
        } else {
            asm volatile("s_wait_asynccnt 0x0" ::: "memory");
        }
        __syncthreads();

        v8f acc0 = {0.f,0.f,0.f,0.f,0.f,0.f,0.f,0.f};
        v8f acc1 = acc0, acc2 = acc0;

        const float* arow = &sx[cur][mrow][0];
        for (int k = 0; k < IN; k += 4) {
            v2f a  = *(const v2f*)(arow + k + kh);   // ds_load_b64
            v2f bA = *(const v2f*)(w0 + k + kh);
            v2f bB = *(const v2f*)(w1 + k + kh);
            v2f bC = *(const v2f*)(w2 + k + kh);
            acc0 = wmma_f32(a, bA, acc0);
            acc1 = wmma_f32(a, bB, acc1);
            acc2 = wmma_f32(a, bC, acc2);
        }

        const long mt0  = (long)(t0 + it) * 16;
        const int rbase = (lane >> 4) * 8;
#pragma unroll
        for (int i = 0; i < 3; ++i) {
            const int col = n0 + i * 16 + mrow;
            const float bias = b_ih[col];
            v8f acc = (i == 0) ? acc0 : (i == 1) ? acc1 : acc2;
#pragma unroll
            for (int r = 0; r < 8; ++r) {
                long row = mt0 + rbase + r;
                xg[row * GATES + col] = acc[r] + bias;
            }
        }
        __syncthreads();   // buffer[cur] free for reuse at it+2
    }
}

// ---------------------------------------------------------------------------
// Kernel 2: sequential GRU scan over SEQ steps, fused online-softmax attention
// and final FC. Grid: 16 blocks (one per 16-batch-row chunk), 128 threads.
// w_hh B-fragments live in registers for the whole scan; h lives in LDS.
// xg gate inputs for step t+1 are prefetched during step t's WMMA work.
// ---------------------------------------------------------------------------
__global__ __launch_bounds__(128) void gru_scan_attn(
    const float* __restrict__ xg, const float* __restrict__ w_hh,
    const float* __restrict__ b_hh, const float* __restrict__ w_attn,
    const float* __restrict__ b_attn, const float* __restrict__ w_fc,
    const float* __restrict__ b_fc, float* __restrict__ out)
{
    __shared__ float sh[16][HID];      // hidden state h_t for 16 batch rows
    __shared__ float shg[16][GATES];   // h @ w_hh^T
    __shared__ float sctx[16][HID];    // running softmax-weighted context
    __shared__ float swa[HID];         // w_attn
    __shared__ float sbh[GATES];       // b_hh
    __shared__ float srow_m[16], srow_d[16], srow_s[16], srow_p[16];

    const int tid  = threadIdx.x;
    const int lane = tid & 31;
    const int wave = tid >> 5;
    const int mrow = lane & 15;
    const int kh   = (lane >> 4) * 2;
    const int b0   = blockIdx.x * 16;

    for (int e = tid; e < 16 * HID; e += 128) {
        (&sh[0][0])[e]   = 0.f;
        (&sctx[0][0])[e] = 0.f;
    }
    for (int e = tid; e < HID;   e += 128) swa[e] = w_attn[e];
    for (int e = tid; e < GATES; e += 128) sbh[e] = b_hh[e];
    if (tid < 16) { srow_m[tid] = -INFINITY; srow_d[tid] = 0.f; }

    // Preload w_hh B-fragments into registers: wave owns N-tiles wave*3..+2.
    v2f bfrag[3][16];
    const int n0 = wave * 48;
#pragma unroll
    for (int i = 0; i < 3; ++i) {
        const float* wrow = w_hh + (n0 + i * 16 + mrow) * HID;
#pragma unroll
        for (int s = 0; s < 16; ++s)
            bfrag[i][s] = *(const v2f*)(wrow + 4 * s + kh);
    }
    __syncthreads();

    for (int t = 0; t < SEQ; ++t) {
        // Prefetch next step's gate inputs (96 cache lines of 128B) while the
        // matrix work below runs.
        if (t + 1 < SEQ && tid < 96) {
            const int m = tid / 6, l = tid % 6;
            const float* pa =
                xg + ((long)(b0 + m) * SEQ + (t + 1)) * GATES + l * 32;
            __builtin_prefetch(pa, 0, 3);
        }

        // ---- 1) hg = h @ w_hh^T via 48 f32 WMMAs per wave ----
        v8f acc0 = {0.f,0.f,0.f,0.f,0.f,0.f,0.f,0.f};
        v8f acc1 = acc0, acc2 = acc0;
#pragma unroll
        for (int s = 0; s < 16; ++s) {
            v2f a = *(const v2f*)(&sh[mrow][4 * s + kh]);
            acc0 = wmma_f32(a, bfrag[0][s], acc0);
            acc1 = wmma_f32(a, bfrag[1][s], acc1);
            acc2 = wmma_f32(a, bfrag[2][s], acc2);
        }
        const int rbase = (lane >> 4) * 8;
#pragma unroll
        for (int i = 0; i < 3; ++i) {
            const int col = n0 + i * 16 + mrow;
            v8f acc = (i == 0) ? acc0 : (i == 1) ? acc1 : acc2;
#pragma unroll
            for (int r = 0; r < 8; ++r)
                shg[rbase + r][col] = acc[r];
        }
        __syncthreads();

        // ---- 2) gate math: 1024 elems, 8 per thread ----
#pragma unroll
        for (int u = 0; u < 8; ++u) {
            const int e = tid + u * 128;
            const int m = e >> 6, j = e & 63;
            const long rb = ((long)(b0 + m) * SEQ + t) * GATES;
            const float xr = xg[rb + j];
            const float xz = xg[rb + 64 + j];
            const float xn = xg[rb + 128 + j];
            const float hr = shg[m][j]       + sbh[j];
            const float hz = shg[m][64 + j]  + sbh[64 + j];
            const float hn = shg[m][128 + j] + sbh[128 + j];
            const float r = 1.f / (1.f + __expf(-(xr + hr)));
            const float z = 1.f / (1.f + __expf(-(xz + hz)));
            const float n = tanhf(xn + r * hn);
            sh[m][j] = (1.f - z) * n + z * sh[m][j];
        }
        __syncthreads();

        // ---- 3) attention logit + online softmax state (one thread per row) --
        if (tid < 16) {
            float dotv = b_attn[0];
#pragma unroll
            for (int j = 0; j < HID; ++j) dotv += sh[tid][j] * swa[j];
            const float mo = srow_m[tid];
            const float mn = fmaxf(mo, dotv);
            const float sc = __expf(mo - mn);
            const float p  = __expf(dotv - mn);
            srow_d[tid] = srow_d[tid] * sc + p;
            srow_m[tid] = mn;
            srow_s[tid] = sc;
            srow_p[tid] = p;
        }
        __syncthreads();

        // ---- 4) running context update ----
#pragma unroll
        for (int u = 0; u < 8; ++u) {
            const int e = tid + u * 128;
            const int m = e >> 6, j = e & 63;
            sctx[m][j] = sctx[m][j] * srow_s[m] + srow_p[m] * sh[m][j];
        }
        __syncthreads();
    }

    // ---- final FC: out[b][c] = (ctx[b]/d[b]) . w_fc[c] + b_fc[c] ----
    if (tid < 32) {
        const int m = tid & 15, c = tid >> 4;
        const float inv_d = 1.f / srow_d[m];
        float acc = b_fc[c];
        for (int j = 0; j < HID; ++j)
            acc += (sctx[m][j] * inv_d) * w_fc[c * HID + j];
        out[(b0 + m) * 2 + c] = acc;
    }
}

// ---------------------------------------------------------------------------
extern "C" void kernel_launch(void* const* d_in, const int* in_sizes, int n_in,
                              void* d_out, int out_size, void* d_ws, size_t ws_size,
                              hipStream_t stream) {
    (void)in_sizes; (void)n_in; (void)out_size; (void)ws_size;
    const float* x      = (const float*)d_in[0];
    const float* w_ih   = (const float*)d_in[1];
    const float* w_hh   = (const float*)d_in[2];
    const float* b_ih   = (const float*)d_in[3];
    const float* b_hh   = (const float*)d_in[4];
    const float* w_attn = (const float*)d_in[5];
    const float* b_attn = (const float*)d_in[6];
    const float* w_fc   = (const float*)d_in[7];
    const float* b_fc   = (const float*)d_in[8];
    float* out = (float*)d_out;
    float* xg  = (float*)d_ws;   // [B*S, 192] = 96 MB scratch

    const int mtiles = (BATCH * SEQ) / 16;                    // 8192
    gru_input_gemm<<<mtiles / TILES_PER_BLOCK, 128, 0, stream>>>(x, w_ih, b_ih,
                                                                 xg);
    gru_scan_attn<<<BATCH / 16, 128, 0, stream>>>(xg, w_hh, b_hh, w_attn, b_attn,
                                                  w_fc, b_fc, out);
}